// GPF_Pool_40853728920209
// MI455X (gfx1250) — compile-verified
//
#include <hip/hip_runtime.h>

#define EMB   1024
#define NKEYS 65536
#define TOPK  512
#define BATCH 128

typedef float v2f __attribute__((ext_vector_type(2)));
typedef float v4f __attribute__((ext_vector_type(4)));
typedef float v8f __attribute__((ext_vector_type(8)));

// ---------------------------------------------------------------------------
// Kernel 1: cosine-ranking scores via V_WMMA_F32_16X16X4_F32.
// One wave handles 16 rows of `keys`. A-matrix layout (32-bit A 16x4):
//   lanes 0-15  : VGPR0=K0, VGPR1=K1   -> float2 load at keys[row][k+0]
//   lanes 16-31 : VGPR0=K2, VGPR1=K3   -> float2 load at keys[row][k+2]
// B-matrix (4x16) is query broadcast across all 16 columns:
//   VGPR0: lanes0-15=q[k]   lanes16-31=q[k+1]
//   VGPR1: lanes0-15=q[k+2] lanes16-31=q[k+3]
// C/D 16x16 f32: VGPR r, lanes 0-15 -> M=r ; lanes 16-31 -> M=r+8.
// q_norm is a rank-invariant positive scale and is dropped.
// ---------------------------------------------------------------------------
__global__ __launch_bounds__(256) void scores_wmma(const float* __restrict__ keys,
                                                   const float* __restrict__ query,
                                                   float* __restrict__ scores) {
  const int lane = threadIdx.x & 31;
  const int wave = threadIdx.x >> 5;
  const int half = lane >> 4;      // 0: K0/K1 half, 1: K2/K3 half
  const int r15  = lane & 15;      // row within 16-row tile
  const int rowbase = (blockIdx.x * 8 + wave) * 16;
  const float* krow = keys + (size_t)(rowbase + r15) * EMB + half * 2;

  v8f c = {0.f, 0.f, 0.f, 0.f, 0.f, 0.f, 0.f, 0.f};
  float sq = 0.f;

#pragma unroll 4
  for (int k = 0; k < EMB; k += 4) {
    v2f a = *(const v2f*)(krow + k);                 // this lane's 2 A elements
    float4 q = *(const float4*)(query + k);          // wave-uniform -> s_load
    v2f b;
    b[0] = half ? q.y : q.x;
    b[1] = half ? q.w : q.z;
    sq = fmaf(a[0], a[0], sq);
    sq = fmaf(a[1], a[1], sq);
    c = __builtin_amdgcn_wmma_f32_16x16x4_f32(false, a, false, b,
                                              (short)0, c, false, false);
  }

  // ||keys[row]||^2 : lane pair (l, l+16) covers the same row.
  float sqFull = sq + __shfl_xor(sq, 16);
  float nrm = fmaxf(sqrtf(sqFull), 1e-8f);

  // Emit: row m<8 -> lane m holds dot in c[m]; row m>=8 -> lane m+16 holds c[m-8].
#pragma unroll
  for (int r = 0; r < 8; ++r) {
    int writer = half ? (r + 24) : r;
    if (lane == writer) {
      int m = half ? (r + 8) : r;
      scores[rowbase + m] = c[r] / nrm;
    }
  }
}

// ---------------------------------------------------------------------------
// Kernel 2: exact sorted top-512 of 65536 scores. Single 1024-thread WG.
// MSB-first 8-bit radix select (LDS histogram) finds threshold key T, then
// candidates (key >= T) are gathered and bitonic-sorted descending on
// (key << 32) | ~index  => value desc, index asc (JAX tie-break).
// ---------------------------------------------------------------------------
__device__ inline unsigned monoKey(float f) {
  unsigned u = __float_as_uint(f);
  return (u & 0x80000000u) ? ~u : (u | 0x80000000u);  // monotonic: bigger f -> bigger u
}

__global__ __launch_bounds__(1024) void radix_topk(const float* __restrict__ scores,
                                                   int* __restrict__ topk) {
  __shared__ unsigned hist[256];
  __shared__ int s_bin;
  __shared__ unsigned s_gt;
  __shared__ unsigned s_cnt;
  __shared__ unsigned long long cand[1024];

  const int tid = threadIdx.x;
  unsigned prefix = 0;
  int target = TOPK;

  for (int shift = 24; shift >= 0; shift -= 8) {
    if (tid < 256) hist[tid] = 0u;
    __syncthreads();
    for (int i = tid; i < NKEYS; i += 1024) {
      unsigned key = monoKey(scores[i]);
      bool match = (shift == 24) || ((key >> (shift + 8)) == prefix);
      if (match) atomicAdd(&hist[(key >> shift) & 255], 1u);
    }
    __syncthreads();
    if (tid == 0) {
      unsigned cum = 0;
      int b = 255;
      for (; b > 0; --b) {
        if (cum + hist[b] >= (unsigned)target) break;
        cum += hist[b];
      }
      s_bin = b;
      s_gt = cum;  // count strictly above chosen bin
    }
    __syncthreads();
    prefix = (prefix << 8) | (unsigned)s_bin;
    target -= (int)s_gt;
    __syncthreads();
  }
  const unsigned T = prefix;  // 512th-largest key

  if (tid == 0) s_cnt = 0u;
  __syncthreads();
  for (int i = tid; i < NKEYS; i += 1024) {
    unsigned key = monoKey(scores[i]);
    if (key >= T) {
      unsigned p = atomicAdd(&s_cnt, 1u);
      if (p < 1024u)
        cand[p] = ((unsigned long long)key << 32) | (unsigned)(~i);
    }
  }
  __syncthreads();
  unsigned n = s_cnt;
  n = n < 1024u ? n : 1024u;
  if ((unsigned)tid >= n) cand[tid] = 0ull;  // pad (key=0 == -inf, below any real T)
  __syncthreads();

  // Bitonic sort, 1024 elements, descending.
  for (int k2 = 2; k2 <= 1024; k2 <<= 1) {
    for (int j = k2 >> 1; j > 0; j >>= 1) {
      int ixj = tid ^ j;
      if (ixj > tid) {
        unsigned long long a = cand[tid], b = cand[ixj];
        bool up = (tid & k2) == 0;
        bool doSwap = up ? (a < b) : (a > b);
        if (doSwap) { cand[tid] = b; cand[ixj] = a; }
      }
      __syncthreads();
    }
  }

  if (tid < TOPK) topk[tid] = (int)(~(unsigned)(cand[tid] & 0xFFFFFFFFull));
}

// ---------------------------------------------------------------------------
// Kernel 3: out[b,j,:] = x[b,j,:] + prompts[topk[j],:]
// x/out are pure 512MB streams with zero reuse -> non-temporal (TH=NT) so the
// stream does not evict the 2MB of gathered prompt rows (re-read 128x) from
// the 192MB L2; prompts keep default RT hint.
// ---------------------------------------------------------------------------
__global__ __launch_bounds__(256) void add_gather(const float* __restrict__ x,
                                                  const float* __restrict__ prompts,
                                                  const int* __restrict__ topk,
                                                  float* __restrict__ out) {
  const int j = blockIdx.x;   // rank slot 0..511
  const int b = blockIdx.y;   // batch 0..127
  const int t = threadIdx.x;  // 256 threads x float4 = 1024 floats
  const int idx = topk[j];    // uniform -> scalar load
  const size_t xoff = ((size_t)b * TOPK + j) * EMB + (size_t)t * 4;

  v4f xa = __builtin_nontemporal_load((const v4f*)(x + xoff));
  v4f p  = *(const v4f*)(prompts + (size_t)idx * EMB + (size_t)t * 4);
  v4f o = xa + p;
  __builtin_nontemporal_store(o, (v4f*)(out + xoff));
}

extern "C" void kernel_launch(void* const* d_in, const int* in_sizes, int n_in,
                              void* d_out, int out_size, void* d_ws, size_t ws_size,
                              hipStream_t stream) {
  const float* x       = (const float*)d_in[0];
  const float* query   = (const float*)d_in[1];
  const float* keys    = (const float*)d_in[2];
  const float* prompts = (const float*)d_in[3];
  float* out = (float*)d_out;

  float* scores = (float*)d_ws;                                        // 65536 f32
  int*   topk   = (int*)((char*)d_ws + (size_t)NKEYS * sizeof(float)); // 512 i32

  // 16 rows/wave, 8 waves/block -> 128 rows/block -> 512 blocks.
  scores_wmma<<<NKEYS / 128, 256, 0, stream>>>(keys, query, scores);
  radix_topk<<<1, 1024, 0, stream>>>(scores, topk);
  dim3 grid(TOPK, BATCH);
  add_gather<<<grid, 256, 0, stream>>>(x, prompts, topk, out);
}